// Parity_9603546874313
// MI455X (gfx1250) — compile-verified
//
#include <hip/hip_runtime.h>
#include <stdint.h>

// Parity/poly-features kernel for MI455X (gfx1250), memory-bound streaming.
// out[b, t] = x[b, i_t] * x[b, j_t]   (j_t == 24 -> sentinel 1.0f)
//   t in [0,24): singles (i=t, j=24)
//   t in [24,300): pairs (i<j), lexicographic - matches itertools.combinations
//
// Roofline: ~6.3 MB read + ~78.6 MB write at 23.3 TB/s => ~3.7 us floor;
// ~20M multiplies is negligible. Pure store-bandwidth-bound, so we use the
// CDNA5 async global->LDS path for input staging and NT b128 stores for the
// streamed output. WMMA is deliberately NOT used: any matmul formulation
// (one-hot gather GEMMs) adds ~2 GFLOP of f32 WMMA and breaks store
// coalescing via the 16x16 C/D lane layout - strictly slower than the
// memory floor.

#define N_BITS      24
#define N_TERMS     300
#define ROWS        32              // batch rows staged per block
#define THREADS     256             // 8 wave32 waves
#define ROW_STRIDE  25              // 24 features + one 1.0f sentinel slot
#define CHUNKS_PER_ROW (N_TERMS / 4)   // 75 float4 chunks per output row

typedef __attribute__((ext_vector_type(4))) float v4f;  // native vector for NT store

struct PairTab { uint16_t v[N_TERMS]; };

static constexpr PairTab makeTab() {
    PairTab t{};
    int n = 0;
    for (int i = 0; i < N_BITS; ++i)                    // singles: x[i] * 1.0
        t.v[n++] = (uint16_t)(i | (N_BITS << 8));
    for (int i = 0; i < N_BITS; ++i)                    // pairs: x[i] * x[j], i<j
        for (int j = i + 1; j < N_BITS; ++j)
            t.v[n++] = (uint16_t)(i | (j << 8));
    return t;
}

__constant__ __align__(8) PairTab c_tab = makeTab();

__global__ __launch_bounds__(THREADS)
void parity_poly2_kernel(const float* __restrict__ in, float* __restrict__ out) {
    __shared__ float smem[ROWS * ROW_STRIDE];   // 3200 B

    const int tid = threadIdx.x;
    const int blk = blockIdx.x;

    // LDS byte address of smem (generic-pointer low 32 bits == wave LDS offset,
    // which is what the async-load VDST operand expects).
    const uint32_t lds_base = (uint32_t)(uintptr_t)(&smem[0]);

    // ---- Stage input tile: 32 rows x 24 floats = 192 x 16B async chunks ----
    if (tid < ROWS * 6) {
        const int r = tid / 6;                 // row within tile
        const int q = tid - r * 6;             // 16B chunk within row (24 floats = 6 chunks)
        const float* gptr = in + ((size_t)blk * ROWS + r) * N_BITS + q * 4;
        const uint32_t laddr = lds_base + (uint32_t)(r * (ROW_STRIDE * 4) + q * 16);
        asm volatile("global_load_async_to_lds_b128 %0, %1, off"
                     :: "v"(laddr), "v"(gptr)
                     : "memory");
    } else if (tid < ROWS * 6 + ROWS) {
        // plant the 1.0f sentinel at slot 24 of each LDS row
        smem[(tid - ROWS * 6) * ROW_STRIDE + N_BITS] = 1.0f;
    }

    asm volatile("s_wait_asynccnt 0x0" ::: "memory");
    __syncthreads();

    // ---- Emit 32 rows x 300 terms as coalesced non-temporal b128 stores ----
    const int total_chunks = ROWS * CHUNKS_PER_ROW;    // 2400
    for (int c = tid; c < total_chunks; c += THREADS) {
        const int r  = c / CHUNKS_PER_ROW;
        const int q  = c - r * CHUNKS_PER_ROW;
        const int t0 = q * 4;

        // 4 packed (i,j) byte-pairs in one 8-byte constant load
        const uint64_t pk = *(const uint64_t*)(&c_tab.v[t0]);
        const float* row = &smem[r * ROW_STRIDE];

        v4f o;
        o.x = row[(pk      ) & 31] * row[(pk >> 8 ) & 31];
        o.y = row[(pk >> 16) & 31] * row[(pk >> 24) & 31];
        o.z = row[(pk >> 32) & 31] * row[(pk >> 40) & 31];
        o.w = row[(pk >> 48) & 31] * row[(pk >> 56) & 31];

        v4f* dst = (v4f*)(out + ((size_t)blk * ROWS + r) * N_TERMS + t0);
        __builtin_nontemporal_store(o, dst);   // 78.6 MB streamed once; bypass cache retention
    }
}

extern "C" void kernel_launch(void* const* d_in, const int* in_sizes, int n_in,
                              void* d_out, int out_size, void* d_ws, size_t ws_size,
                              hipStream_t stream) {
    const float* in  = (const float*)d_in[0];
    float*       out = (float*)d_out;

    const int batch = in_sizes[0] / N_BITS;    // 65536
    const int grid  = batch / ROWS;            // 2048 blocks of 8 wave32 waves

    parity_poly2_kernel<<<dim3(grid), dim3(THREADS), 0, stream>>>(in, out);
}